// HarmonicOscillator_70291434766378
// MI455X (gfx1250) — compile-verified
//
#include <hip/hip_runtime.h>

// HarmonicOscillator on MI455X (gfx1250, wave32).
//   f0:[8,1,500] cn:[8,100,500] an:[8,1,500] -> out:[8,40000] (all f32)
// Structure:
//  * Hann-OLA upsample collapses to 2 taps: win[80-r]*x[j] + win[160-r]*x[j+1]
//  * phi_h = h * (2pi/fs) * cumsum(f0_up) -> one scan/batch + rotation recurrence
//  * Nyquist mask monotone in h -> early break (~41/100 active on average)
//  * online softmax fused with sin-weighted sum (single pass over h)
//  * cn tile [100 rows x 2 cols] (row stride 500 floats) staged to LDS via TDM
//    tensor_load_to_lds + s_wait_tensorcnt (CDNA5 async-tensor path)

#define B_    8
#define NH_   100
#define T_    500      // frames
#define F_    80       // framesize
#define TOUT_ (T_ * F_)
#define FS_   16000.0f
#define NYQ_  8000.0f
#define TWO_PI_ 6.283185307179586f
#define BLK_  96       // 3 waves; lanes 0..79 produce output

typedef unsigned int v4u __attribute__((ext_vector_type(4)));
typedef int          v8i __attribute__((ext_vector_type(8)));
typedef int          v4i __attribute__((ext_vector_type(4)));

#if defined(__HIP_DEVICE_COMPILE__) && __has_builtin(__builtin_amdgcn_tensor_load_to_lds)
#define USE_TDM_ 1
#else
#define USE_TDM_ 0
#endif

__device__ __forceinline__ float f0_up_val(const float* __restrict__ f0b, int n) {
  float pos = ((float)n + 0.5f) * (1.0f / (float)F_) - 0.5f;
  pos = fminf(fmaxf(pos, 0.0f), (float)(T_ - 1));
  float fi0 = floorf(pos);
  int i0 = (int)fi0;
  int i1 = min(i0 + 1, T_ - 1);
  float w = pos - fi0;
  return f0b[i0] * (1.0f - w) + f0b[i1] * w;
}

struct Smem {
  float cn[2 * NH_];   // TDM destination: row h -> {cn[b,h,j], cn[b,h,j+1]}; first => LDS offset 0
  float scan[BLK_];
};

// Main kernel: one block per (batch, frame). TDM issued immediately by wave 0.
__global__ void __launch_bounds__(BLK_) ho_main(const float* __restrict__ f0,
                                                const float* __restrict__ cn,
                                                const float* __restrict__ an,
                                                const float* __restrict__ S,
                                                float* __restrict__ out) {
  __shared__ Smem sm;
  const int blk = blockIdx.x;
  const int b = blk / T_;
  const int j = blk - b * T_;
  const int tid = threadIdx.x;

#if USE_TDM_
  if (tid < 32) {  // wave 0 issues the DMA (wave-coherent branch)
    unsigned lds_off = (unsigned)(unsigned long long)(&sm.cn[0]);  // LDS aperture: low 32 bits = byte offset
    unsigned long long ga =
        (unsigned long long)(const void*)(cn + ((size_t)b * NH_ * T_ + (size_t)j));
    // D# group0: count=1, lds_addr, global_addr[56:0], type=2
    v4u g0 = { 1u, lds_off, (unsigned)(ga & 0xffffffffu),
               (unsigned)((ga >> 32) & 0x01ffffffu) | (2u << 30) };
    const int rem = T_ - j;  // dim0 elems remaining from tile start -> HW zero-fills col j+1 at edge
    // D# group1: data_size=4B; tensor_dim0=rem, tensor_dim1=NH; tile 2 x NH; row stride = T_
    v8i g1 = { 0x00020000,
               (rem & 0xffff) << 16,          // tensor_dim0[15:0] in bits[63:48]
               (NH_ & 0xffff) << 16,          // tensor_dim0[31:16]=0 | tensor_dim1[15:0]
               (2 << 16),                     // tensor_dim1[31:16]=0 | tile_dim0=2
               NH_,                           // tile_dim1=NH, tile_dim2=0
               T_,                            // tensor_dim0_stride low 32
               0, 0 };
    v4i gz = {0, 0, 0, 0};
#if __clang_major__ >= 23
    v8i gz8 = {0, 0, 0, 0, 0, 0, 0, 0};
    __builtin_amdgcn_tensor_load_to_lds(g0, g1, gz, gz, gz8, 0);
#else
    __builtin_amdgcn_tensor_load_to_lds(g0, g1, gz, gz, 0);
#endif
#if __has_builtin(__builtin_amdgcn_s_wait_tensorcnt)
    __builtin_amdgcn_s_wait_tensorcnt(0);
#else
    asm volatile("s_wait_tensorcnt 0" ::: "memory");
#endif
  }
#else
  // Cooperative strided fill of the cn tile (host pass / no-TDM toolchains).
  for (int idx = tid; idx < 2 * NH_; idx += BLK_) {
    int h = idx >> 1, xcol = idx & 1;
    int col = j + xcol;
    sm.cn[idx] = (col < T_) ? cn[((size_t)b * NH_ + h) * T_ + col] : 0.0f;
  }
#endif

  // f0_up at this output sample + intra-frame inclusive scan (phase).
  const float* f0b = f0 + b * T_;
  const int n = j * F_ + tid;
  float u = (tid < F_) ? f0_up_val(f0b, n) : 0.0f;
  sm.scan[tid] = u;
  __syncthreads();
  for (int off = 1; off < BLK_; off <<= 1) {   // offsets 1..64 cover 96 entries
    float add = (tid >= off) ? sm.scan[tid - off] : 0.0f;
    __syncthreads();
    sm.scan[tid] += add;
    __syncthreads();
  }
  if (tid >= F_) return;  // after final barrier

  float x = S[b * T_ + j] + sm.scan[tid];   // cumsum(f0_up) up to n, inclusive
  x -= FS_ * floorf(x * (1.0f / FS_));      // phase periodic in fs -> range-reduce
  float phi = x * (TWO_PI_ / FS_);
  float s1, c1;
  __sincosf(phi, &s1, &c1);

  const int r = tid;
  float wA = 0.5f * (1.0f - __cosf((float)(F_ - r) * (TWO_PI_ / 161.0f)));
  float wB = 0.5f * (1.0f - __cosf((float)(2 * F_ - r) * (TWO_PI_ / 161.0f)));

  float an0 = an[b * T_ + j];
  float an1 = (j + 1 < T_) ? an[b * T_ + j + 1] : 0.0f;
  float anv = wA * an0 + wB * an1;

  // Fused: online softmax over active harmonics + sin(h*phi) via rotation.
  float sh = s1, ch = c1;                 // sin/cos of h*phi, h=1
  float m = -3.0e38f, den = 0.0f, num = 0.0f;
  for (int h = 1; h <= NH_; ++h) {
    if ((float)h * u >= NYQ_) break;      // mask monotone in h
    float v = wA * sm.cn[2 * (h - 1)] + wB * sm.cn[2 * (h - 1) + 1];
    if (v > m) {
      float scale = __expf(m - v);        // first hit: exp(-huge)=0, resets cleanly
      den *= scale; num *= scale; m = v;
    }
    float e = __expf(v - m);
    den += e;
    num = __fmaf_rn(e, sh, num);
    float sn = __fmaf_rn(sh, c1, ch * s1);
    ch = __fmaf_rn(ch, c1, -sh * s1);
    sh = sn;
  }
  out[(size_t)b * TOUT_ + n] = anv * num / den;
}

// Per-batch exclusive prefix sums of f0_up at frame boundaries.
// One block per batch, 512 threads; thread j sums its frame's 80 interp values,
// then a Hillis-Steele scan in LDS.
__global__ void __launch_bounds__(512) ho_framescan(const float* __restrict__ f0,
                                                    float* __restrict__ S) {
  int b = blockIdx.x;
  int tid = threadIdx.x;
  const float* f0b = f0 + b * T_;
  float fsum = 0.0f;
  if (tid < T_) {
    int base = tid * F_;
#pragma unroll 4
    for (int r = 0; r < F_; ++r) fsum += f0_up_val(f0b, base + r);
  }
  __shared__ float sc[512];
  sc[tid] = fsum;
  __syncthreads();
  for (int off = 1; off < 512; off <<= 1) {
    float add = (tid >= off) ? sc[tid - off] : 0.0f;
    __syncthreads();
    sc[tid] += add;
    __syncthreads();
  }
  if (tid < T_) S[b * T_ + tid] = sc[tid] - fsum;  // exclusive prefix
}

extern "C" void kernel_launch(void* const* d_in, const int* in_sizes, int n_in,
                              void* d_out, int out_size, void* d_ws, size_t ws_size,
                              hipStream_t stream) {
  (void)in_sizes; (void)n_in; (void)out_size; (void)ws_size;
  const float* f0 = (const float*)d_in[0];
  const float* cn = (const float*)d_in[1];
  const float* an = (const float*)d_in[2];
  float* out = (float*)d_out;
  float* S = (float*)d_ws;  // B_*T_ floats = 16 KB scratch

  hipLaunchKernelGGL(ho_framescan, dim3(B_), dim3(512), 0, stream, f0, S);
  hipLaunchKernelGGL(ho_main, dim3(B_ * T_), dim3(BLK_), 0, stream, f0, cn, an, S, out);
}